// proposal_clloss_75685913690507
// MI455X (gfx1250) — compile-verified
//
#include <hip/hip_runtime.h>
#include <hip/hip_bf16.h>
#include <stdint.h>

typedef __attribute__((ext_vector_type(2))) float v2f;
typedef __attribute__((ext_vector_type(8))) float v8f;
typedef __attribute__((ext_vector_type(4))) unsigned int u32x4;
typedef __attribute__((ext_vector_type(8))) unsigned int u32x8;

#define BS    32
#define CH    128
#define GN    256
#define SS    16
#define NG    (BS * GN)
#define FROW  (SS + 1)      // 17-float LDS row stride (TDM pad) -> conflict-free
#define TINV  5.0f          // 1/T, T = 0.2
#define WLOSS 0.1f
#define BIGL  1073741824    // 2^30 sentinel

__global__ __launch_bounds__(32)
void proposal_clloss_group_kernel(const int*   __restrict__ labs,
                                  const float* __restrict__ gfeat,
                                  const int*   __restrict__ gidx,
                                  const float* __restrict__ ctx,
                                  float*       __restrict__ accum)
{
    // TDM deposits the [128 x 16] f32 tile row-by-row; pad_enable inserts
    // 1 DWORD after every 16 DWORDs -> effective row stride 17 floats.
    __shared__ __align__(16) float featCS[CH * FROW];   // [c][s] padded
    __shared__ float Msh[SS][CH + 1];                   // instance-mean rows
    __shared__ float simS[SS][SS + 1];
    __shared__ int   labS[SS];
    __shared__ int   idxS[SS];
    __shared__ float rowLi[SS];

    const int lane = threadIdx.x;
    const int g    = blockIdx.x;
    const int b    = g / GN;
    const int gn   = g % GN;

    // ---- kick off the TDM DMA of the feature tile (global -> LDS) --------
    // Tile: tile_dim0 = 16 elements (one channel's samples, contiguous),
    // tile_dim1 = 128 rows, row stride tensor_dim0_stride = Gn*S = 4096 elems.
    {
        const uint64_t gaddr  = (uint64_t)(uintptr_t)(gfeat
                                + (size_t)b * CH * GN * SS + (size_t)gn * SS);
        const uint32_t ldsOff = (uint32_t)(uintptr_t)(void*)featCS;

        u32x4 g0;
        g0.x = 1u;                                        // count=1 (valid D#)
        g0.y = ldsOff;                                    // lds_addr
        g0.z = (uint32_t)gaddr;                           // global_addr[31:0]
        g0.w = ((uint32_t)(gaddr >> 32) & 0x01FFFFFFu)    // global_addr[56:32]
               | 0x80000000u;                             // type = 2 ("image")

        u32x8 g1;
        g1.s0 = (2u << 16)          // data_size = 2 -> 4 bytes
              | (1u << 20)          // pad_enable
              | (3u << 22);         // pad_interval = 3 -> every 16 DWORDs
                                    // pad_amount  = 0 -> 1 DWORD
        g1.s1 = (uint32_t)SS << 16; // tensor_dim0 = 16 (bits[63:48] = low16)
        g1.s2 = (uint32_t)CH << 16; // tensor_dim1 = 128 (bits[95:80] = low16)
        g1.s3 = (uint32_t)SS << 16; // tile_dim0 = 16 (bits[127:112])
        g1.s4 = (uint32_t)CH;       // tile_dim1 = 128; tile_dim2 = 0
        g1.s5 = (uint32_t)(GN * SS);// tensor_dim0_stride = 4096 (low32)
        g1.s6 = 0u;                 // stride high bits / dim1_stride low = 0
        g1.s7 = 0u;

        u32x4 gz = {0u, 0u, 0u, 0u};
        asm volatile("tensor_load_to_lds %0, %1, %2, %3"
                     :: "s"(g0), "s"(g1), "s"(gz), "s"(gz)
                     : "memory");
    }

    if (lane < SS) {
        labS[lane]  = labs[g * SS + lane];
        idxS[lane]  = gidx[g * SS + lane];
        rowLi[lane] = 0.0f;
    }
    __syncthreads();

    // ---- redundant (wave-uniform) membership computation, S=16 is tiny ----
    int u = 0;
    for (int t = 0; t < SS; ++t) {
        bool first = true;
        for (int j = 0; j < t; ++j) first &= (idxS[j] != idxS[t]);
        u += first ? 1 : 0;
    }
    int labm[SS];
    for (int t = 0; t < SS; ++t) labm[t] = (t < u) ? labS[t] : BIGL;

    unsigned bgMask = 0u;
    for (int t = 0; t < SS; ++t) if (labm[t] == -1) bgMask |= (1u << t);
    const bool hasBg = (bgMask != 0u);

    int      nFg = 0;
    unsigned fgMask[SS];
    float    fgCnt[SS];
    for (int t = 0; t < SS; ++t) {
        const int L = labm[t];
        if (L == BIGL || L == -1) continue;
        bool first = true;
        for (int j = 0; j < t; ++j) first &= (labm[j] != L);
        if (first && nFg < SS - 1) {
            unsigned m = 0u; int cnt = 0;
            for (int s2 = 0; s2 < SS; ++s2)
                if (labm[s2] == L) { m |= (1u << s2); ++cnt; }
            fgMask[nFg] = m; fgCnt[nFg] = (float)cnt; ++nFg;
        }
    }
    const int instNum = nFg + 1;

    // ---- wait for the TDM tile, then build instance-mean rows M[slot][c] --
    __builtin_amdgcn_s_wait_tensorcnt(0);

    for (int j = 0; j < CH / 32; ++j) {
        const int c = lane + 32 * j;
        const float* fr = &featCS[c * FROW];
        float v;
        if (hasBg) {
            float sum = 0.0f, n = 0.0f;
            for (int t = 0; t < SS; ++t)
                if ((bgMask >> t) & 1u) { sum += fr[t]; n += 1.0f; }
            v = sum / n;
        } else {
            v = ctx[c];
        }
        Msh[0][c] = v;
        for (int k = 0; k < nFg; ++k) {
            float sum = 0.0f;
            const unsigned m = fgMask[k];
            for (int t = 0; t < SS; ++t)
                if ((m >> t) & 1u) sum += fr[t];
            Msh[1 + k][c] = sum / fgCnt[k];
        }
        for (int k = instNum; k < SS; ++k) Msh[k][c] = 0.0f;
    }
    __syncthreads();

    // ---- Gram matrix sim = M * M^T via V_WMMA_F32_16X16X4_F32, K=128 ----
    // A 16x4 f32 fragment: lane<16 -> {M[lane][k0], M[lane][k0+1]},
    // lane>=16 -> {M[lane-16][k0+2], M[lane-16][k0+3]}. For the symmetric
    // product, the B (4x16) fragment is register-identical, so pass A twice.
    v8f acc = {};
    const int r    = lane & (SS - 1);
    const int koff = (lane >> 4) << 1;
    #pragma unroll
    for (int kb = 0; kb < CH / 4; ++kb) {
        const int k0 = kb * 4 + koff;
        v2f a;
        a.x = Msh[r][k0];
        a.y = Msh[r][k0 + 1];
        acc = __builtin_amdgcn_wmma_f32_16x16x4_f32(
                  /*neg_a=*/false, a, /*neg_b=*/false, a,
                  /*c_mod=*/(short)0, acc, /*reuse_a=*/false, /*reuse_b=*/false);
    }

    // C/D layout: VGPR v holds row (v + 8*(lane>=16)), col = lane&15
    {
        const int col   = lane & (SS - 1);
        const int rbase = (lane >> 4) * 8;
        #pragma unroll
        for (int v = 0; v < 8; ++v)
            simS[rbase + v][col] = acc[v] * TINV;
    }
    __syncthreads();

    // ---- masked row-wise logsumexp minus diagonal, fg rows only ----
    if (lane < SS) {
        float li = 0.0f;
        if (lane >= 1 && lane < instNum) {
            float mx = -__builtin_inff();
            for (int j2 = 0; j2 < instNum; ++j2) mx = fmaxf(mx, simS[lane][j2]);
            float se = 0.0f;
            for (int j2 = 0; j2 < instNum; ++j2) se += __expf(simS[lane][j2] - mx);
            const float lse = mx + __logf(se);
            li = lse - simS[lane][lane];
        }
        rowLi[lane] = li;
    }
    __syncthreads();

    if (lane == 0 && nFg >= 1) {
        float gl = 0.0f;
        for (int t = 1; t < SS; ++t) gl += rowLi[t];
        gl /= (float)(instNum - 1);
        atomicAdd(&accum[0], gl);     // sum(gl * gv)
        atomicAdd(&accum[1], 1.0f);   // sum(gv)
    }
}

__global__ void proposal_clloss_init_kernel(float* accum)
{
    if (threadIdx.x < 2) accum[threadIdx.x] = 0.0f;
}

__global__ void proposal_clloss_final_kernel(const float* __restrict__ accum,
                                             float* __restrict__ out)
{
    if (threadIdx.x == 0)
        out[0] = (accum[1] > 0.0f) ? (accum[0] / accum[1]) * WLOSS : 0.0f;
}

extern "C" void kernel_launch(void* const* d_in, const int* in_sizes, int n_in,
                              void* d_out, int out_size, void* d_ws, size_t ws_size,
                              hipStream_t stream)
{
    const int*   labs  = (const int*)  d_in[0];  // proposal_instance_mask [bs,Gn,S]
    const float* gfeat = (const float*)d_in[1];  // grouped_features [bs,C,Gn,S]
    const int*   gidx  = (const int*)  d_in[2];  // grouped_indices [bs,Gn,S]
    const float* ctx   = (const float*)d_in[3];  // context_compen [1,C]
    float* accum = (float*)d_ws;                 // [0]=sum(gl*gv), [1]=sum(gv)
    float* out   = (float*)d_out;

    proposal_clloss_init_kernel<<<1, 32, 0, stream>>>(accum);
    proposal_clloss_group_kernel<<<NG, 32, 0, stream>>>(labs, gfeat, gidx, ctx, accum);
    proposal_clloss_final_kernel<<<1, 32, 0, stream>>>(accum, out);
}